// RBF_74526272520773
// MI455X (gfx1250) — compile-verified
//
#include <hip/hip_runtime.h>
#include <hip/hip_bf16.h>

typedef float v2f __attribute__((ext_vector_type(2)));
typedef float v8f __attribute__((ext_vector_type(8)));

#define DDIM 64      // feature dimension
#define TILE 128     // output tile edge (per workgroup)
#define NKER 5       // number of bandwidth multipliers

// ---------------------------------------------------------------------------
// K1: per-row squared norms + per-block partial sum of squared norms.
// One wave (32 lanes) per row; block = 256 threads = 8 rows.
// ---------------------------------------------------------------------------
__global__ void rbf_rownorms(const float* __restrict__ X,
                             float* __restrict__ sqn,
                             float* __restrict__ partS1) {
    const int wave = threadIdx.x >> 5;
    const int lane = threadIdx.x & 31;
    const int row  = blockIdx.x * 8 + wave;

    float2 x = *(const float2*)(X + (size_t)row * DDIM + lane * 2);
    float  s = x.x * x.x + x.y * x.y;
    #pragma unroll
    for (int off = 16; off > 0; off >>= 1)
        s += __shfl_xor(s, off, 32);

    __shared__ float red[8];
    if (lane == 0) { sqn[row] = s; red[wave] = s; }
    __syncthreads();
    if (threadIdx.x == 0) {
        float t = 0.f;
        #pragma unroll
        for (int i = 0; i < 8; ++i) t += red[i];
        partS1[blockIdx.x] = t;
    }
}

// ---------------------------------------------------------------------------
// K2: single block. Column sums of X (for the analytic sum(L2)), reduce the
// S1 partials, compute bw and write -1/(bw*mult[k]) for the exp epilogue.
// sum(L2) = 2*N*sum_i||x_i||^2 - 2*||sum_i x_i||^2   (clamp term negligible)
// ---------------------------------------------------------------------------
__global__ void rbf_finalize(const float* __restrict__ X,
                             const float* __restrict__ partS1,
                             const float* __restrict__ mult,
                             float* __restrict__ negInv,
                             int N, int nPart) {
    __shared__ float cs[4 * DDIM];
    __shared__ float s1red[256];

    const int t   = threadIdx.x;
    const int col = t & 63;
    const int grp = t >> 6;

    float c = 0.f;
    for (int r = grp; r < N; r += 4) c += X[(size_t)r * DDIM + col];
    cs[grp * DDIM + col] = c;

    float s1 = 0.f;
    for (int i = t; i < nPart; i += 256) s1 += partS1[i];
    s1red[t] = s1;
    __syncthreads();

    if (t < DDIM) {
        float tot = cs[col] + cs[DDIM + col] + cs[2 * DDIM + col] + cs[3 * DDIM + col];
        cs[col] = tot * tot;
    }
    __syncthreads();

    if (t == 0) {
        float S1 = 0.f;
        for (int i = 0; i < 256; ++i) S1 += s1red[i];
        float csq = 0.f;
        for (int k = 0; k < DDIM; ++k) csq += cs[k];
        double sumL2 = 2.0 * (double)N * (double)S1 - 2.0 * (double)csq;
        float  bw    = (float)(sumL2 / ((double)N * (double)N - (double)N));
        for (int k = 0; k < NKER; ++k)
            negInv[k] = -1.0f / (bw * mult[k]);
    }
}

// ---------------------------------------------------------------------------
// K3: fused Gram(WMMA f32 16x16x4) + RBF-sum epilogue.
// 128x128 tile per block, 8 waves, each wave = 32x64 (2x4 WMMA tiles).
// ---------------------------------------------------------------------------
__global__ __launch_bounds__(256, 2)
void rbf_wmma(const float* __restrict__ X,
              const float* __restrict__ sqn,
              const float* __restrict__ negInv,
              float* __restrict__ out, int N) {
    __shared__ float As[TILE * DDIM];   // rows m0..m0+127 of X
    __shared__ float Bs[TILE * DDIM];   // rows n0..n0+127 of X
    const int m0  = blockIdx.y * TILE;
    const int n0  = blockIdx.x * TILE;
    const int tid = threadIdx.x;

    // Cooperative load: 128 rows x 16 float4 per panel.
    #pragma unroll
    for (int i = 0; i < 8; ++i) {
        int idx = tid + i * 256;        // 0..2047
        int r   = idx >> 4;             // 0..127
        int c4  = idx & 15;             // float4 column
        float4 a = *(const float4*)(X + (size_t)(m0 + r) * DDIM + c4 * 4);
        float4 b = *(const float4*)(X + (size_t)(n0 + r) * DDIM + c4 * 4);
        *(float4*)(As + r * DDIM + c4 * 4) = a;
        *(float4*)(Bs + r * DDIM + c4 * 4) = b;
    }
    __syncthreads();

    const int wave  = tid >> 5;
    const int lane  = tid & 31;
    const int wm    = (wave >> 1) * 32;   // wave M offset within tile
    const int wn    = (wave & 1) * 64;    // wave N offset within tile
    const int mfrag = lane & 15;          // row index within 16x16 fragment
    const int kh    = (lane >> 4) * 2;    // K sub-offset (0 or 2)

    v8f acc[2][4] = {};

    // 16 K-steps of V_WMMA_F32_16X16X4_F32  -> 128 WMMAs per wave
    #pragma unroll
    for (int k0 = 0; k0 < DDIM; k0 += 4) {
        v2f a[2], b[4];
        #pragma unroll
        for (int i = 0; i < 2; ++i)
            a[i] = *(const v2f*)(As + (wm + i * 16 + mfrag) * DDIM + k0 + kh);
        #pragma unroll
        for (int j = 0; j < 4; ++j)
            b[j] = *(const v2f*)(Bs + (wn + j * 16 + mfrag) * DDIM + k0 + kh);
        #pragma unroll
        for (int i = 0; i < 2; ++i)
            #pragma unroll
            for (int j = 0; j < 4; ++j)
                acc[i][j] = __builtin_amdgcn_wmma_f32_16x16x4_f32(
                    false, a[i], false, b[j], (short)0, acc[i][j], false, false);
    }

    // Epilogue: L2 = max(||xi||^2 + ||xj||^2 - 2 g, 0); out = sum_k exp(-L2/scale_k)
    float ns[NKER];
    #pragma unroll
    for (int k = 0; k < NKER; ++k) ns[k] = negInv[k];

    #pragma unroll
    for (int i = 0; i < 2; ++i) {
        #pragma unroll
        for (int j = 0; j < 4; ++j) {
            int   n    = n0 + wn + j * 16 + (lane & 15);
            float sn_n = sqn[n];
            #pragma unroll
            for (int v = 0; v < 8; ++v) {
                int   m  = m0 + wm + i * 16 + (lane >> 4) * 8 + v;
                float g  = acc[i][j][v];
                float L2 = fmaxf(sqn[m] + sn_n - 2.0f * g, 0.0f);
                float o  = 0.0f;
                #pragma unroll
                for (int k = 0; k < NKER; ++k)
                    o += __expf(L2 * ns[k]);
                out[(size_t)m * N + n] = o;
            }
        }
    }
}

// ---------------------------------------------------------------------------
extern "C" void kernel_launch(void* const* d_in, const int* in_sizes, int n_in,
                              void* d_out, int out_size, void* d_ws, size_t ws_size,
                              hipStream_t stream) {
    const float* X    = (const float*)d_in[0];
    const float* mult = (const float*)d_in[1];
    float*       out  = (float*)d_out;

    const int N  = in_sizes[0] / DDIM;   // 4096
    const int nb = N / 8;                // K1 blocks (one wave per row)

    float* sqn    = (float*)d_ws;        // N floats
    float* partS1 = sqn + N;             // nb floats
    float* negInv = partS1 + nb;         // NKER floats

    rbf_rownorms<<<nb, 256, 0, stream>>>(X, sqn, partS1);
    rbf_finalize<<<1, 256, 0, stream>>>(X, partS1, mult, negInv, N, nb);

    dim3 grid(N / TILE, N / TILE);       // 32 x 32 tiles
    rbf_wmma<<<grid, 256, 0, stream>>>(X, sqn, negInv, out, N);
}